// LabelLoss_67611375174145
// MI455X (gfx1250) — compile-verified
//
#include <hip/hip_runtime.h>
#include <hip/hip_bf16.h>

#ifndef __has_builtin
#define __has_builtin(x) 0
#endif

// CDNA5 async global->LDS DMA path (ASYNCcnt). Guarded: falls back to direct
// vectorized global loads if this toolchain doesn't expose the builtins.
#if defined(__HIP_DEVICE_COMPILE__) && \
    __has_builtin(__builtin_amdgcn_global_load_async_to_lds_b128) && \
    __has_builtin(__builtin_amdgcn_s_wait_asynccnt)
#define USE_ASYNC_LDS 1
#else
#define USE_ASYNC_LDS 0
#endif

namespace {
constexpr int kB = 64, kJ = 32, kM = 160, kN = 160;
constexpr int kHW = kM * kN;                         // 25600 floats per (b,j)
constexpr int kThreads = 256;                        // 8 x wave32
constexpr int kTile = 5120;                          // 20 KB LDS tile
constexpr int kNumTiles = kHW / kTile;               // 5
constexpr int kV4PerTile = kTile / 4;                // 1280
constexpr int kV4PerThread = kV4PerTile / kThreads;  // 5 b128 loads/thread/tile

// Heatmap = 200 MiB vs 192 MB global L2: pin ~156 MiB (1600 pairs) with the
// HT temporal hint so it survives graph replays; stream the rest NT so it
// never evicts the resident set.
constexpr int kResidentPairs = 1600;
// gfx12 CPol: TH in bits [2:0]. Load TH: 1 = NT, 2 = HT.
constexpr int kCpolHT = 2;
constexpr int kCpolNT = 1;
}

#if USE_ASYNC_LDS
typedef int v4i __attribute__((vector_size(16)));
typedef __attribute__((address_space(1))) v4i as1_v4i;  // global
typedef __attribute__((address_space(3))) v4i as3_v4i;  // LDS
#endif

__device__ __forceinline__ void upd(float v, int ei, float& best, int& bidx) {
  if (v > best) { best = v; bidx = ei; }  // strict '>' keeps first-max per thread
}

#if USE_ASYNC_LDS
// Double-buffered async-DMA argmax scan of one (b,j) heatmap slice.
// CPOL is the temporal hint applied to the streaming loads.
template <int CPOL>
__device__ __forceinline__ void scan_pair_async(const float* __restrict__ src,
                                                float (*buf)[kTile], int tid,
                                                float& best, int& bidx) {
  // Prefetch tile 0 (5 x b128 per thread -> ASYNCcnt += 5 per wave).
  {
    float* dst = &buf[0][0];
    #pragma unroll
    for (int w = 0; w < kV4PerThread; ++w) {
      const int slot = w * kThreads + tid;
      __builtin_amdgcn_global_load_async_to_lds_b128(
          (as1_v4i*)(src + slot * 4), (as3_v4i*)(dst + slot * 4), 0, CPOL);
    }
  }
  for (int t = 0; t < kNumTiles; ++t) {
    if (t + 1 < kNumTiles) {
      // Stream next tile while we scan the current one.
      const float* s = src + (t + 1) * kTile;
      float* dst = &buf[(t + 1) & 1][0];
      #pragma unroll
      for (int w = 0; w < kV4PerThread; ++w) {
        const int slot = w * kThreads + tid;
        __builtin_amdgcn_global_load_async_to_lds_b128(
            (as1_v4i*)(s + slot * 4), (as3_v4i*)(dst + slot * 4), 0, CPOL);
      }
      __builtin_amdgcn_s_wait_asynccnt(5);  // async loads retire in order:
                                            // <=5 pending => tile t done (this wave)
    } else {
      __builtin_amdgcn_s_wait_asynccnt(0);
    }
    __syncthreads();  // tile t resident for all 8 waves
    const float4* tb = (const float4*)&buf[t & 1][0];
    const int base = t * kTile;
    #pragma unroll
    for (int w = 0; w < kV4PerThread; ++w) {
      const int slot = w * kThreads + tid;
      const float4 v = tb[slot];
      const int ei = base + slot * 4;
      upd(v.x, ei,     best, bidx);
      upd(v.y, ei + 1, best, bidx);
      upd(v.z, ei + 2, best, bidx);
      upd(v.w, ei + 3, best, bidx);
    }
    __syncthreads();  // everyone done reading buf[t&1] before tile t+2 overwrites it
  }
}
#endif

__global__ __launch_bounds__(kThreads)
void labelloss_argmax_kernel(const float* __restrict__ pred,
                             const float* __restrict__ gt,
                             const float* __restrict__ heatmap,
                             float* __restrict__ per_obj) {
  const int pair = blockIdx.x;  // pair = b*kJ + j
  const int tid  = threadIdx.x;
  const float* __restrict__ src = heatmap + (size_t)pair * kHW;

  float best = -__builtin_inff();
  int   bidx = 0;

#if USE_ASYNC_LDS
  __shared__ __align__(16) float buf[2][kTile];  // 40 KB double buffer
  if (pair < kResidentPairs) {
    scan_pair_async<kCpolHT>(src, buf, tid, best, bidx);  // keep hot in L2
  } else {
    scan_pair_async<kCpolNT>(src, buf, tid, best, bidx);  // pure stream
  }
#else
  // Fallback: direct coalesced b128 loads (still HBM-rate limited).
  const float4* __restrict__ s4 = (const float4*)src;
  #pragma unroll 5
  for (int w = 0; w < (kHW / 4) / kThreads; ++w) {
    const int slot = w * kThreads + tid;
    const float4 v = s4[slot];
    const int ei = slot * 4;
    upd(v.x, ei,     best, bidx);
    upd(v.y, ei + 1, best, bidx);
    upd(v.z, ei + 2, best, bidx);
    upd(v.w, ei + 3, best, bidx);
  }
#endif

  // wave32 argmax reduction; tie -> smaller index (jnp first-max semantics).
  #pragma unroll
  for (int m = 16; m > 0; m >>= 1) {
    const float ov = __shfl_xor(best, m, 32);
    const int   oi = __shfl_xor(bidx, m, 32);
    if (ov > best || (ov == best && oi < bidx)) { best = ov; bidx = oi; }
  }

  __shared__ float wv[kThreads / 32];
  __shared__ int   wi[kThreads / 32];
  if ((tid & 31) == 0) { wv[tid >> 5] = best; wi[tid >> 5] = bidx; }
  __syncthreads();

  if (tid == 0) {
    #pragma unroll
    for (int w = 1; w < kThreads / 32; ++w) {
      if (wv[w] > best || (wv[w] == best && wi[w] < bidx)) { best = wv[w]; bidx = wi[w]; }
    }
    const int b = pair / kJ, j = pair % kJ;
    const float x = (float)(bidx / kM);   // idx // m
    const float y = (float)(bidx % kM);   // idx %  m
    const float* __restrict__ g = gt + (size_t)pair * 11;            // gt[b][j][*]
    const float* __restrict__ p = pred + (size_t)b * 9 * kJ + j;     // pred[b][k][j][0] = p[k*kJ]
    const float g9 = g[9], g10 = g[10];
    const bool valid = (g9 > 0.f) && (g10 > 0.f) && (g9 < (float)kM) && (g10 < (float)kN);
    const float dx = g9  + g[7] - x - p[7 * kJ];
    const float dy = g10 + g[8] - y - p[8 * kJ];
    float loss = dx * dx + dy * dy;
    #pragma unroll
    for (int k = 0; k < 7; ++k) {
      const float d = p[k * kJ] - g[k];
      loss += d * d;
    }
    per_obj[pair] = valid ? loss : 0.f;
  }
}

// Deterministic per-batch sum over j (no float atomics -> bitwise stable
// across graph replays).
__global__ void labelloss_sum_kernel(const float* __restrict__ per_obj,
                                     float* __restrict__ out) {
  const int b = threadIdx.x;
  if (b < kB) {
    float s = 0.f;
    #pragma unroll
    for (int j = 0; j < kJ; ++j) s += per_obj[b * kJ + j];
    out[b] = s;
  }
}

extern "C" void kernel_launch(void* const* d_in, const int* in_sizes, int n_in,
                              void* d_out, int out_size, void* d_ws, size_t ws_size,
                              hipStream_t stream) {
  const float* pred    = (const float*)d_in[0];  // (64, 9, 32, 1) f32
  const float* gt      = (const float*)d_in[1];  // (64, 32, 11)   f32
  const float* heatmap = (const float*)d_in[2];  // (64, 32, 160, 160) f32
  float* out     = (float*)d_out;                // (64,) f32
  float* per_obj = (float*)d_ws;                 // 2048 floats of scratch

  labelloss_argmax_kernel<<<kB * kJ, kThreads, 0, stream>>>(pred, gt, heatmap, per_obj);
  labelloss_sum_kernel<<<1, 64, 0, stream>>>(per_obj, out);
}